// OneEuro_4870492914168
// MI455X (gfx1250) — compile-verified
//
#include <hip/hip_runtime.h>
#include <hip/hip_bf16.h>
#include <math.h>

#define AS1 __attribute__((address_space(1)))
#define AS3 __attribute__((address_space(3)))

typedef __attribute__((ext_vector_type(4))) int v4i;

namespace {
constexpr int B    = 32;
constexpr int T    = 4096;
constexpr int C    = 512;
constexpr int CPB  = 256;            // channels per block == blockDim.x
constexpr int CT   = 256;            // time steps produced per block
constexpr int HALO = 32;             // warm-up steps (error ~0.137^32 ~ 1e-27)
constexpr int TT   = 16;             // time rows per LDS tile
constexpr int TILE_ELEMS = TT * CPB; // 4096 floats = 16 KB
constexpr int NBUF = 3;              // triple buffer: 2 tiles in flight

constexpr double TWO_PI = 6.283185307179586476925286766559;
constexpr float AD   = (float)(TWO_PI / (TWO_PI + 1.0));       // a_d (fs=1, dcutoff=1)
constexpr float OMAD = (float)(1.0 - TWO_PI / (TWO_PI + 1.0)); // 1 - a_d
constexpr float RC0  = (float)TWO_PI;                          // 2*pi*MIN_CUTOFF
constexpr float RC1  = (float)(TWO_PI * 0.007);                // 2*pi*BETA
} // namespace

#if __has_builtin(__builtin_amdgcn_global_load_async_to_lds_b128)
#define ONEEURO_ASYNC 1
#endif

#if __has_builtin(__builtin_amdgcn_s_wait_asynccnt)
#define WAIT_ASYNC(n) __builtin_amdgcn_s_wait_asynccnt(n)
#else
#define WAIT_ASYNC(n) asm volatile("s_wait_asynccnt %0" ::"i"(n) : "memory")
#endif

__device__ __forceinline__ float fast_rcp(float v) {
#if __has_builtin(__builtin_amdgcn_rcpf)
  return __builtin_amdgcn_rcpf(v);
#else
  return 1.0f / v;
#endif
}

#if defined(ONEEURO_ASYNC)
// Issue async copies for one 16x256-float tile: 4 passes, each pass all 256
// lanes move 16B (4 rows of 1KB per pass).  Tracked on ASYNCcnt.
__device__ __forceinline__ void issue_tile(const float* gtile, float* ltile,
                                           int lane) {
  const int r4  = lane >> 6; // 0..3 : row within pass
  const int seg = lane & 63; // 0..63: 16B segment within 1KB row
  const float* g = gtile + (size_t)r4 * C + seg * 4;
  float* l = ltile + r4 * CPB + seg * 4;
#pragma unroll
  for (int p = 0; p < 4; ++p) {
    __builtin_amdgcn_global_load_async_to_lds_b128((AS1 v4i*)(void*)g,
                                                   (AS3 v4i*)(void*)l, 0, 0);
    g += 4 * C;
    l += 4 * CPB;
  }
}
#endif

__global__ __launch_bounds__(CPB) void OneEuro_kernel(
    const float* __restrict__ x, float* __restrict__ y) {
  const int cl = threadIdx.x;      // channel lane within slab
  const int c0 = blockIdx.x * CPB; // channel slab base (0 or 256)
  const int kc = blockIdx.y;       // time chunk index
  const int b  = blockIdx.z;       // batch

  const int t0   = kc * CT;
  const int halo = (kc == 0) ? 0 : HALO;
  const int ts   = t0 - halo;

  const size_t plane = (size_t)b * T * C + c0;
  float* __restrict__ outp = y + plane + cl;

  int t = ts;
  float xp = 0.f, dxp = 0.f, yp = 0.f;
  bool started = false;

#if defined(ONEEURO_ASYNC)
  const int ntiles = (CT + halo) / TT;                 // 16 or 18
  const float* __restrict__ gchunk = x + plane + (size_t)ts * C;
  __shared__ float smem[NBUF * TILE_ELEMS];            // 48 KB triple buffer

  // Prologue: put two tiles in flight.
  issue_tile(gchunk, smem, cl);
  issue_tile(gchunk + (size_t)TT * C, smem + TILE_ELEMS, cl);

  int rd = 0;          // buffer holding tile `tile`
  int wr = 2;          // buffer to fill with tile `tile+2`
  for (int tile = 0; tile < ntiles; ++tile) {
    const int remain = ntiles - 1 - tile;
    if (remain >= 2)
      issue_tile(gchunk + (size_t)(tile + 2) * TT * C, smem + wr * TILE_ELEMS,
                 cl);
    // Loads retire in order: waiting to <= 4*min(2,remain) outstanding batches
    // guarantees tile `tile`'s 4 issues have landed in LDS.
    if (remain >= 2)      WAIT_ASYNC(8);
    else if (remain == 1) WAIT_ASYNC(4);
    else                  WAIT_ASYNC(0);
    __syncthreads(); // all waves' async writes to this buffer visible
    const float* buf = smem + rd * TILE_ELEMS;
#pragma unroll
    for (int i = 0; i < TT; ++i) {
      float xc = buf[i * CPB + cl];
      if (!started) {
        yp = xc; xp = xc; started = true;
        if (t >= t0) outp[(size_t)t * C] = yp; // y[0] = x[0]
      } else {
        float dx  = xc - xp;                    // fs == 1
        float dxh = fmaf(AD, dx, OMAD * dxp);   // derivative EMA
        float r   = fmaf(RC1, fabsf(dxh), RC0); // 2*pi*cutoff
        float inv = fast_rcp(r + 1.0f);         // = 1 - a_x
        float ax  = 1.0f - inv;
        float yt  = fmaf(ax, xc, inv * yp);
        if (t >= t0) outp[(size_t)t * C] = yt;
        xp = xc; dxp = dxh; yp = yt;
      }
      ++t;
    }
    __syncthreads(); // buffer fully consumed before it is refilled
    rd = (rd == NBUF - 1) ? 0 : rd + 1;
    wr = (wr == NBUF - 1) ? 0 : wr + 1;
  }
#else
  // Fallback: direct coalesced streaming loads with global_prefetch_b8.
  const float* __restrict__ inp = x + plane + cl;
  const int steps = CT + halo;
  for (int s = 0; s < steps; ++s) {
    if (s + 8 < steps)
      __builtin_prefetch(inp + (size_t)(t + 8) * C, 0, 0);
    float xc = inp[(size_t)t * C];
    if (!started) {
      yp = xc; xp = xc; started = true;
      if (t >= t0) outp[(size_t)t * C] = yp;
    } else {
      float dx  = xc - xp;
      float dxh = fmaf(AD, dx, OMAD * dxp);
      float r   = fmaf(RC1, fabsf(dxh), RC0);
      float inv = fast_rcp(r + 1.0f);
      float ax  = 1.0f - inv;
      float yt  = fmaf(ax, xc, inv * yp);
      if (t >= t0) outp[(size_t)t * C] = yt;
      xp = xc; dxp = dxh; yp = yt;
    }
    ++t;
  }
#endif
}

extern "C" void kernel_launch(void* const* d_in, const int* in_sizes, int n_in,
                              void* d_out, int out_size, void* d_ws,
                              size_t ws_size, hipStream_t stream) {
  (void)in_sizes; (void)n_in; (void)out_size; (void)d_ws; (void)ws_size;
  const float* x = (const float*)d_in[0];
  float* y = (float*)d_out;
  dim3 grid(C / CPB, T / CT, B); // (2, 16, 32) = 1024 blocks
  dim3 block(CPB);               // 256 threads = 8 waves (wave32)
  hipLaunchKernelGGL(OneEuro_kernel, grid, block, 0, stream, x, y);
}